// Spatial_attn_17205638987881
// MI455X (gfx1250) — compile-verified
//
#include <hip/hip_runtime.h>
#include <math.h>

// Problem dims (fixed by reference)
constexpr int NB = 32;   // batch
constexpr int NN = 512;  // nodes
constexpr int NS = 64;   // seq len / hidden width
constexpr int SC = 8;    // s-dimension split of the score reduction

typedef float v2f __attribute__((ext_vector_type(2)));
typedef float v8f __attribute__((ext_vector_type(8)));

// ---------------------------------------------------------------------------
// Once: TH = tanh(hidden)  [B,S,N], and transpose global_inputs -> GIT [S,B,N]
// so the per-step gate reads are fully coalesced.
// ---------------------------------------------------------------------------
__global__ __launch_bounds__(256)
void precompute_kernel(const float* __restrict__ hidden,  // [B,S,N]
                       const float* __restrict__ gin,     // [B,N,S]
                       float* __restrict__ TH,            // [B,S,N]
                       float* __restrict__ GIT)           // [S,B,N]
{
    int idx = blockIdx.x * 256 + threadIdx.x;
    if (idx >= NB * NS * NN) return;
    TH[idx] = tanhf(hidden[idx]);
    // idx enumerates (b, n, t) with t fastest -> coalesced read of gin
    int t = idx % NS;
    int n = (idx / NS) % NN;
    int b = idx / (NS * NN);
    GIT[(t * NB + b) * NN + n] = gin[idx];
}

// ---------------------------------------------------------------------------
// Per step: x = attn * inp_t, written straight into the output slice (which
// also serves as the GEMM's A matrix).
// ---------------------------------------------------------------------------
__global__ __launch_bounds__(256)
void gate_kernel(const float* __restrict__ attn,   // [B,N]
                 const float* __restrict__ git_t,  // [B,N] slice of GIT
                 float* __restrict__ x_out)        // d_out + t*B*N
{
    int idx = blockIdx.x * 256 + threadIdx.x;
    if (idx < NB * NN) x_out[idx] = attn[idx] * git_t[idx];
}

// ---------------------------------------------------------------------------
// Per step: y = X @ W + b  ([32,512]x[512,64]) in fp32 via V_WMMA_F32_16X16X4,
// then TY = tanh(y). One 1024-thread workgroup = 32 waves:
//   8 output tiles (2 M-tiles x 4 N-tiles) x 4 K-slices of 128.
// Each wave WMMA-accumulates its K-slice; partial tiles are combined in LDS
// (deterministic fixed-order sum), then bias + tanh.
// A frag (16x4 f32, ISA 7.12.2): lanes 0-15 hold K=k0,k0+1; lanes 16-31 K=k0+2,k0+3.
// C/D (16x16 f32): VGPR j = row mt*16 + (lane<16 ? j : 8+j), col nt*16 + lane%16.
// ---------------------------------------------------------------------------
__global__ __launch_bounds__(1024)
void gemm_tanh_kernel(const float* __restrict__ X,    // [B,N] (current x)
                      const float* __restrict__ W,    // [N,S]
                      const float* __restrict__ bias, // [S]
                      float* __restrict__ TY)         // [B,S] = tanh(y)
{
    __shared__ float lds[32 * 256];   // per-wave 16x16 partial tiles (32 KB)

    int wave = threadIdx.x >> 5;     // 0..31
    int lane = threadIdx.x & 31;
    int kt   = wave & 3;             // K slice 0..3
    int tile = wave >> 2;            // 0..7
    int mt   = tile >> 2;            // 0..1  (M tile)
    int nt   = tile & 3;             // 0..3  (N tile)
    int half = lane >> 4;            // 0 or 1 (lane half)
    int l16  = lane & 15;

    int arow = mt * 16 + l16;        // A row (M) for this lane
    int bcol = nt * 16 + l16;        // B col (N) for this lane

    v8f c = {};
    int kbeg = kt * (NN / 4);
    for (int k0 = kbeg; k0 < kbeg + NN / 4; k0 += 4) {
        int ka = k0 + (half << 1);   // K base for this lane half
        v2f a, bb;
        a.x  = X[arow * NN + ka + 0];
        a.y  = X[arow * NN + ka + 1];
        bb.x = W[(ka + 0) * NS + bcol];
        bb.y = W[(ka + 1) * NS + bcol];
        c = __builtin_amdgcn_wmma_f32_16x16x4_f32(
                /*neg_a=*/false, a, /*neg_b=*/false, bb,
                /*c_mod=*/(short)0, c, /*reuse_a=*/false, /*reuse_b=*/false);
    }

    // stash this wave's partial tile
    #pragma unroll
    for (int j = 0; j < 8; ++j) lds[wave * 256 + j * 32 + lane] = c[j];
    __syncthreads();

    // one wave per tile (kt==0) combines the 4 K-slice partials
    if (kt == 0) {
        int mbase = mt * 16 + (half << 3);
        float bv  = bias[bcol];
        #pragma unroll
        for (int j = 0; j < 8; ++j) {
            float y = c[j]
                    + lds[(wave + 1) * 256 + j * 32 + lane]
                    + lds[(wave + 2) * 256 + j * 32 + lane]
                    + lds[(wave + 3) * 256 + j * 32 + lane]
                    + bv;
            TY[(mbase + j) * NS + bcol] = tanhf(y);
        }
    }
}

// ---------------------------------------------------------------------------
// Per step (the hot kernel), stage 1:
//   s[b,n] = sum_{s,k} v[k] * tanh(hidden + y)
//          = sum_{s,k} v[k] * (TH + ty[k]) / (1 + TH*ty[k])   (exact identity)
// Split over SC=8 s-chunks for occupancy: grid (B, SC) = 256 blocks x 512 thr
// = 128K threads. Each thread: 8 TH values register-tiled, 64 k per value;
// per element: add + fma + rcp + mul + fma -> 4 VALU + 1 trans, matching the
// 4:1 VALU:transcendental issue rate so both pipes stay saturated.
// Deterministic: fixed split, fixed-order stage-2 sum (no float atomics).
// ---------------------------------------------------------------------------
__global__ __launch_bounds__(512)
void score_partial_kernel(const float* __restrict__ TH,   // [B,S,N]
                          const float* __restrict__ TY,   // [B,S]
                          const float* __restrict__ v,    // [S]
                          float* __restrict__ PART)       // [SC,B,N]
{
    __shared__ float sty[NS];
    __shared__ float sv[NS];

    int b  = blockIdx.x;     // 0..31
    int sc = blockIdx.y;     // 0..7
    int n  = threadIdx.x;    // 0..511
    if (n < NS) { sty[n] = TY[b * NS + n]; sv[n] = v[n]; }
    __syncthreads();

    const float* thp = TH + (size_t)b * NS * NN + (sc * (NS / SC)) * NN + n;
    float u[NS / SC];
    #pragma unroll
    for (int i = 0; i < NS / SC; ++i) u[i] = thp[i * NN];

    float acc = 0.f;
    for (int k = 0; k < NS; ++k) {
        float ty = sty[k];
        float vk = sv[k];
        #pragma unroll
        for (int i = 0; i < NS / SC; ++i) {
            float num = u[i] + ty;
            float den = __builtin_fmaf(u[i], ty, 1.0f);
            acc = __builtin_fmaf(vk * num, __builtin_amdgcn_rcpf(den), acc);
        }
    }
    PART[((size_t)sc * NB + b) * NN + n] = acc;
}

// ---------------------------------------------------------------------------
// Per step, stage 2: fixed-order combine of the SC partials, then softmax
// over the 512 nodes of each batch row.
// ---------------------------------------------------------------------------
__global__ __launch_bounds__(512)
void softmax_kernel(const float* __restrict__ PART,  // [SC,B,N]
                    float* __restrict__ ATTN)        // [B,N]
{
    __shared__ float red[512];

    int b = blockIdx.x;
    int n = threadIdx.x;
    float acc = 0.f;
    #pragma unroll
    for (int sc = 0; sc < SC; ++sc)
        acc += PART[((size_t)sc * NB + b) * NN + n];

    red[n] = acc;
    __syncthreads();
    for (int off = 256; off > 0; off >>= 1) {
        if (n < off) red[n] = fmaxf(red[n], red[n + off]);
        __syncthreads();
    }
    float mx = red[0];
    __syncthreads();
    float e = __builtin_amdgcn_exp2f((acc - mx) * 1.4426950408889634f);
    red[n] = e;
    __syncthreads();
    for (int off = 256; off > 0; off >>= 1) {
        if (n < off) red[n] += red[n + off];
        __syncthreads();
    }
    ATTN[b * NN + n] = e * __builtin_amdgcn_rcpf(red[0]);
}

// ---------------------------------------------------------------------------
// Launch: precompute once, then 64 sequential steps (softmax feeds next step).
// Workspace (floats): TH[1M] | GIT[1M] | TY[2K] | ATTN[16K] | PART[128K]
// ~8.6 MB total.
// ---------------------------------------------------------------------------
extern "C" void kernel_launch(void* const* d_in, const int* in_sizes, int n_in,
                              void* d_out, int out_size, void* d_ws, size_t ws_size,
                              hipStream_t stream) {
    const float* gin   = (const float*)d_in[0]; // global_inputs [B,N,S]
    const float* hid   = (const float*)d_in[1]; // attention_states [B,S,N]
    const float* v     = (const float*)d_in[2]; // [S]
    const float* attn0 = (const float*)d_in[3]; // [B,N]
    const float* W     = (const float*)d_in[4]; // [N,S]
    const float* bias  = (const float*)d_in[5]; // [S]
    float* out = (float*)d_out;                 // outs [S,B,N]

    float* TH   = (float*)d_ws;                 // B*S*N
    float* GIT  = TH   + NB * NS * NN;          // S*B*N
    float* TY   = GIT  + NS * NB * NN;          // B*S
    float* ATTN = TY   + NB * NS;               // B*N
    float* PART = ATTN + NB * NN;               // SC*B*N

    precompute_kernel<<<(NB * NS * NN + 255) / 256, 256, 0, stream>>>(hid, gin, TH, GIT);

    for (int t = 0; t < NS; ++t) {
        const float* attn = (t == 0) ? attn0 : ATTN;
        float* xslice = out + (size_t)t * NB * NN;
        gate_kernel<<<(NB * NN + 255) / 256, 256, 0, stream>>>(attn, GIT + t * NB * NN, xslice);
        if (t == NS - 1) break;  // final softmax is discarded by the scan
        gemm_tanh_kernel<<<1, 1024, 0, stream>>>(xslice, W, bias, TY);
        score_partial_kernel<<<dim3(NB, SC), 512, 0, stream>>>(TH, TY, v, PART);
        softmax_kernel<<<NB, 512, 0, stream>>>(PART, ATTN);
    }
}